// SANLayer_89129161327109
// MI455X (gfx1250) — compile-verified
//
#include <hip/hip_runtime.h>
#include <cstddef>

#define Nn   50000
#define NEe  512000
#define Dd   128
#define Hh   8
#define DHh  16
#define LSTR 132                 // LDS row stride (floats): 128 + 4 pad, conflict-free
#define NBH  ((Nn + 127) / 128)  // 391 node tiles
#define NBE  (NEe / 128)         // 4000 edge tiles (exact)

typedef __attribute__((ext_vector_type(2))) float v2f;
typedef __attribute__((ext_vector_type(8))) float v8f;

// D(16x16,f32) = A(16x4,f32) * B(4x16,f32) + C  — exact fp32 matrix op on CDNA5
__device__ __forceinline__ v8f wmma4(v2f a, v2f b, v8f c) {
  return __builtin_amdgcn_wmma_f32_16x16x4_f32(false, a, false, b, (short)0, c,
                                               false, false);
}

// Wave computes rows [m0, m0+16) x all 128 cols: 8 accumulator tiles, K=128.
// A in LDS row-major:  Al[row*LSTR + k]   -> a = {Al[.., k0+2h], Al[.., k0+2h+1]}
// B in LDS col-major:  Bl[col*LSTR + k]   -> b = {Bl[.., k0+2h], Bl[.., k0+2h+1]}
// Both fragments are contiguous b64 LDS loads; no register shuffles needed.
__device__ __forceinline__ void gemm128(const float* __restrict__ Al,
                                        const float* __restrict__ Bl,
                                        int m0, int mr, int half, v8f (&acc)[8]) {
  for (int k0 = 0; k0 < 128; k0 += 4) {
    const float* ap = Al + (m0 + mr) * LSTR + k0 + 2 * half;
    v2f a; a.x = ap[0]; a.y = ap[1];
#pragma unroll
    for (int t = 0; t < 8; ++t) {
      const float* bp = Bl + (t * 16 + mr) * LSTR + k0 + 2 * half;
      v2f b; b.x = bp[0]; b.y = bp[1];
      acc[t] = wmma4(a, b, acc[t]);
    }
  }
}

// Load a 128x128 fp32 tile (row-guarded, zero-padded) into LDS, row-major, stride LSTR.
__device__ __forceinline__ void ld_tile(float* __restrict__ L,
                                        const float* __restrict__ g,
                                        int row0, int nrows, size_t rstride,
                                        int col0, int tid) {
#pragma unroll
  for (int i = 0; i < 16; ++i) {
    int idx = tid + i * 256;
    int r = idx >> 5, c4 = idx & 31;
    float4 v = make_float4(0.f, 0.f, 0.f, 0.f);
    int gr = row0 + r;
    if (gr < nrows) v = *(const float4*)(g + (size_t)gr * rstride + col0 + c4 * 4);
    *(float4*)(L + r * LSTR + c4 * 4) = v;
  }
}

// Load a 128x128 fp32 weight block TRANSPOSED into LDS: L[n*LSTR + k] = g[(k0+k)*rs + c0+n]
__device__ __forceinline__ void ld_wT(float* __restrict__ L,
                                      const float* __restrict__ g,
                                      int krow0, size_t rstride, int col0, int tid) {
#pragma unroll
  for (int i = 0; i < 16; ++i) {
    int idx = tid + i * 256;
    int k = idx >> 5, c4 = idx & 31;
    float4 v = *(const float4*)(g + (size_t)(krow0 + k) * rstride + col0 + c4 * 4);
    int n = c4 * 4;
    L[(n + 0) * LSTR + k] = v.x;
    L[(n + 1) * LSTR + k] = v.y;
    L[(n + 2) * LSTR + k] = v.z;
    L[(n + 3) * LSTR + k] = v.w;
  }
}

// ---------------- kernel 1: Q/K/V = h @ {Wq,Wk,Wv} ----------------
__global__ void san_qkv(const float* __restrict__ h, const float* __restrict__ Wq,
                        const float* __restrict__ Wk, const float* __restrict__ Wv,
                        float* __restrict__ Qo, float* __restrict__ Ko,
                        float* __restrict__ Vo) {
  __shared__ float Al[128 * LSTR];
  __shared__ float Bl[128 * LSTR];
  const int tid = threadIdx.x;
  const int row0 = blockIdx.x * 128;
  const float* W = (blockIdx.y == 0) ? Wq : ((blockIdx.y == 1) ? Wk : Wv);
  float* O = (blockIdx.y == 0) ? Qo : ((blockIdx.y == 1) ? Ko : Vo);
  ld_tile(Al, h, row0, Nn, Dd, 0, tid);
  ld_wT(Bl, W, 0, Dd, 0, tid);
  __syncthreads();
  const int lane = tid & 31, wave = tid >> 5;
  const int half = lane >> 4, mr = lane & 15, m0 = wave * 16;
  v8f acc[8] = {};
  gemm128(Al, Bl, m0, mr, half, acc);
#pragma unroll
  for (int t = 0; t < 8; ++t) {
    int col = t * 16 + mr;
#pragma unroll
    for (int r = 0; r < 8; ++r) {
      int grow = row0 + m0 + r + 8 * half;
      if (grow < Nn) O[(size_t)grow * Dd + col] = acc[t][r];
    }
  }
}

// ---- kernel 2: Eh = e@We, e2 = e+Eh (+BN partials), scores, scatter ----
__global__ void san_edge(const float* __restrict__ e, const float* __restrict__ We,
                         const float* __restrict__ Qm, const float* __restrict__ Km,
                         const float* __restrict__ Vm, const int* __restrict__ src,
                         const int* __restrict__ dst, float* __restrict__ e2out,
                         float* __restrict__ wV, float* __restrict__ zb,
                         float* __restrict__ sumE, float* __restrict__ sqE) {
  __shared__ float Al[128 * LSTR];   // e tile, overwritten in place with Eh
  __shared__ float Bl[128 * LSTR];   // We (transposed)
  __shared__ float cs[128];
  __shared__ float cq[128];
  const int tid = threadIdx.x;
  const int row0 = blockIdx.x * 128;
  ld_tile(Al, e, row0, NEe, Dd, 0, tid);
  ld_wT(Bl, We, 0, Dd, 0, tid);
  if (tid < 128) { cs[tid] = 0.f; cq[tid] = 0.f; }
  __syncthreads();

  const int lane = tid & 31, wave = tid >> 5;
  const int half = lane >> 4, mr = lane & 15, m0 = wave * 16;
  v8f acc[8] = {};
  gemm128(Al, Bl, m0, mr, half, acc);
  __syncthreads();

  // e2 = e + Eh; keep Eh in LDS (each (row,col) owned by exactly one lane)
#pragma unroll
  for (int t = 0; t < 8; ++t) {
    int col = t * 16 + mr;
    float s = 0.f, q = 0.f;
#pragma unroll
    for (int r = 0; r < 8; ++r) {
      int row = m0 + r + 8 * half;
      float eh = acc[t][r];
      float ev = Al[row * LSTR + col] + eh;
      e2out[(size_t)(row0 + row) * Dd + col] = ev;
      Al[row * LSTR + col] = eh;
      s += ev; q += ev * ev;
    }
    atomicAdd(&cs[col], s);
    atomicAdd(&cq[col], q);
  }
  __syncthreads();
  if (tid < 128) { atomicAdd(&sumE[tid], cs[tid]); atomicAdd(&sqE[tid], cq[tid]); }

  // scores: one (edge, head) pair per thread-slot; Eh read from LDS
#pragma unroll
  for (int i = 0; i < 4; ++i) {
    int idx = tid + i * 256;          // 0..1023 = 128 edges x 8 heads
    int el = idx >> 3, hh = idx & 7;
    int sN = src[row0 + el];
    int dN = dst[row0 + el];
    const float4* kk = (const float4*)(Km + (size_t)sN * Dd + hh * DHh);
    const float4* qq = (const float4*)(Qm + (size_t)dN * Dd + hh * DHh);
    const float4* eh = (const float4*)(Al + el * LSTR + hh * DHh);
    float dot = 0.f;
#pragma unroll
    for (int j = 0; j < 4; ++j) {
      float4 a = kk[j], b = qq[j], c = eh[j];
      dot += a.x * b.x * c.x + a.y * b.y * c.y + a.z * b.z * c.z + a.w * b.w * c.w;
    }
    dot *= 0.25f;                                 // 1/sqrt(DH)
    dot = fminf(5.f, fmaxf(-5.f, dot));
    float sv = expf(dot);
    atomicAdd(zb + (size_t)dN * Hh + hh, sv);
    const float* vv = Vm + (size_t)sN * Dd + hh * DHh;
    float* wvp = wV + (size_t)dN * Dd + hh * DHh;
#pragma unroll
    for (int j = 0; j < DHh; ++j) atomicAdd(wvp + j, vv[j] * sv);
  }
}

// ---- kernel 3: h2 = h + (wV/(z+1e-6)) @ Wo + bo; BN1 partials ----
__global__ void san_attnout(const float* __restrict__ wV, const float* __restrict__ zb,
                            const float* __restrict__ h, const float* __restrict__ Wo,
                            const float* __restrict__ bo, float* __restrict__ h2raw,
                            float* __restrict__ sum1, float* __restrict__ sq1) {
  __shared__ float Al[128 * LSTR];
  __shared__ float Bl[128 * LSTR];
  __shared__ float cs[128];
  __shared__ float cq[128];
  const int tid = threadIdx.x;
  const int row0 = blockIdx.x * 128;
#pragma unroll
  for (int i = 0; i < 16; ++i) {
    int idx = tid + i * 256;
    int r = idx >> 5, c4 = idx & 31;
    float4 v = make_float4(0.f, 0.f, 0.f, 0.f);
    int gr = row0 + r;
    if (gr < Nn) {
      float4 w4 = *(const float4*)(wV + (size_t)gr * Dd + c4 * 4);
      float inv = 1.f / (zb[(size_t)gr * Hh + (c4 >> 2)] + 1e-6f);
      v.x = w4.x * inv; v.y = w4.y * inv; v.z = w4.z * inv; v.w = w4.w * inv;
    }
    *(float4*)(Al + r * LSTR + c4 * 4) = v;
  }
  ld_wT(Bl, Wo, 0, Dd, 0, tid);
  if (tid < 128) { cs[tid] = 0.f; cq[tid] = 0.f; }
  __syncthreads();
  const int lane = tid & 31, wave = tid >> 5;
  const int half = lane >> 4, mr = lane & 15, m0 = wave * 16;
  v8f acc[8] = {};
  gemm128(Al, Bl, m0, mr, half, acc);
#pragma unroll
  for (int t = 0; t < 8; ++t) {
    int col = t * 16 + mr;
    float s = 0.f, q = 0.f;
#pragma unroll
    for (int r = 0; r < 8; ++r) {
      int grow = row0 + m0 + r + 8 * half;
      if (grow < Nn) {
        float val = acc[t][r] + h[(size_t)grow * Dd + col] + bo[col];
        h2raw[(size_t)grow * Dd + col] = val;
        s += val; q += val * val;
      }
    }
    atomicAdd(&cs[col], s);
    atomicAdd(&cq[col], q);
  }
  __syncthreads();
  if (tid < 128) { atomicAdd(&sum1[tid], cs[tid]); atomicAdd(&sq1[tid], cq[tid]); }
}

// ---- kernel 4: (sum,sumsq) -> (mean, rsqrt(var+eps)) in place ----
__global__ void san_finalize(float* sum, float* sq, float count) {
  int c = threadIdx.x;
  float m = sum[c] / count;
  float var = sq[c] / count - m * m;
  sum[c] = m;
  sq[c] = rsqrtf(var + 1e-5f);
}

// ---- kernel 5: h2n = BN1(h2); X1 = relu(h2n @ W1 + b1) ----
__global__ void san_ffn1(const float* __restrict__ h2raw, const float* __restrict__ mean1,
                         const float* __restrict__ inv1, const float* __restrict__ g1h,
                         const float* __restrict__ b1h, const float* __restrict__ W1,
                         const float* __restrict__ b1, float* __restrict__ h2n,
                         float* __restrict__ X1) {
  __shared__ float Al[128 * LSTR];
  __shared__ float Bl[128 * LSTR];
  const int tid = threadIdx.x;
  const int row0 = blockIdx.x * 128;
#pragma unroll
  for (int i = 0; i < 16; ++i) {
    int idx = tid + i * 256;
    int r = idx >> 5, c4 = idx & 31;
    float4 v = make_float4(0.f, 0.f, 0.f, 0.f);
    int gr = row0 + r;
    if (gr < Nn) {
      float4 x = *(const float4*)(h2raw + (size_t)gr * Dd + c4 * 4);
      int c = c4 * 4;
      v.x = g1h[c + 0] * (x.x - mean1[c + 0]) * inv1[c + 0] + b1h[c + 0];
      v.y = g1h[c + 1] * (x.y - mean1[c + 1]) * inv1[c + 1] + b1h[c + 1];
      v.z = g1h[c + 2] * (x.z - mean1[c + 2]) * inv1[c + 2] + b1h[c + 2];
      v.w = g1h[c + 3] * (x.w - mean1[c + 3]) * inv1[c + 3] + b1h[c + 3];
      *(float4*)(h2n + (size_t)gr * Dd + c4 * 4) = v;
    }
    *(float4*)(Al + r * LSTR + c4 * 4) = v;
  }
  const int lane = tid & 31, wave = tid >> 5;
  const int half = lane >> 4, mr = lane & 15, m0 = wave * 16;
  for (int hb = 0; hb < 2; ++hb) {            // two 128-col halves of W1
    __syncthreads();
    ld_wT(Bl, W1, 0, 256, hb * 128, tid);
    __syncthreads();
    v8f acc[8] = {};
    gemm128(Al, Bl, m0, mr, half, acc);
#pragma unroll
    for (int t = 0; t < 8; ++t) {
      int col = hb * 128 + t * 16 + mr;
#pragma unroll
      for (int r = 0; r < 8; ++r) {
        int grow = row0 + m0 + r + 8 * half;
        if (grow < Nn) {
          float val = fmaxf(acc[t][r] + b1[col], 0.f);
          X1[(size_t)grow * 256 + col] = val;
        }
      }
    }
  }
}

// ---- kernel 6: h3raw = h2n + X1 @ W2 + b2 (K=256); BN2 partials ----
__global__ void san_ffn2(const float* __restrict__ X1, const float* __restrict__ W2,
                         const float* __restrict__ b2, const float* __restrict__ h2n,
                         float* __restrict__ h3raw, float* __restrict__ sum2,
                         float* __restrict__ sq2) {
  __shared__ float Al[128 * LSTR];
  __shared__ float Bl[128 * LSTR];
  __shared__ float cs[128];
  __shared__ float cq[128];
  const int tid = threadIdx.x;
  const int row0 = blockIdx.x * 128;
  if (tid < 128) { cs[tid] = 0.f; cq[tid] = 0.f; }
  const int lane = tid & 31, wave = tid >> 5;
  const int half = lane >> 4, mr = lane & 15, m0 = wave * 16;
  v8f acc[8] = {};
  for (int kc = 0; kc < 2; ++kc) {            // two 128-wide K chunks
    __syncthreads();
    ld_tile(Al, X1, row0, Nn, 256, kc * 128, tid);
    ld_wT(Bl, W2, kc * 128, 128, 0, tid);
    __syncthreads();
    gemm128(Al, Bl, m0, mr, half, acc);
  }
#pragma unroll
  for (int t = 0; t < 8; ++t) {
    int col = t * 16 + mr;
    float s = 0.f, q = 0.f;
#pragma unroll
    for (int r = 0; r < 8; ++r) {
      int grow = row0 + m0 + r + 8 * half;
      if (grow < Nn) {
        float val = acc[t][r] + b2[col] + h2n[(size_t)grow * Dd + col];
        h3raw[(size_t)grow * Dd + col] = val;
        s += val; q += val * val;
      }
    }
    atomicAdd(&cs[col], s);
    atomicAdd(&cq[col], q);
  }
  __syncthreads();
  if (tid < 128) { atomicAdd(&sum2[tid], cs[tid]); atomicAdd(&sq2[tid], cq[tid]); }
}

// ---- kernel 7: in-place BN apply ----
__global__ void san_bnapply(float* __restrict__ x, const float* __restrict__ mean,
                            const float* __restrict__ inv, const float* __restrict__ g,
                            const float* __restrict__ b, size_t total) {
  size_t i = (size_t)blockIdx.x * blockDim.x + threadIdx.x;
  size_t stride = (size_t)gridDim.x * blockDim.x;
  for (; i < total; i += stride) {
    int c = (int)(i & 127);
    x[i] = g[c] * (x[i] - mean[c]) * inv[c] + b[c];
  }
}

extern "C" void kernel_launch(void* const* d_in, const int* in_sizes, int n_in,
                              void* d_out, int out_size, void* d_ws, size_t ws_size,
                              hipStream_t stream) {
  (void)in_sizes; (void)n_in; (void)out_size; (void)ws_size;
  const float* h   = (const float*)d_in[0];
  const float* e   = (const float*)d_in[1];
  const int*   src = (const int*)d_in[2];
  const int*   dst = (const int*)d_in[3];
  const float* Wq  = (const float*)d_in[4];
  const float* Wk  = (const float*)d_in[5];
  const float* Wv  = (const float*)d_in[6];
  const float* We  = (const float*)d_in[7];
  const float* Wo  = (const float*)d_in[8];
  const float* bo  = (const float*)d_in[9];
  const float* g1h = (const float*)d_in[10];
  const float* b1h = (const float*)d_in[11];
  const float* g1e = (const float*)d_in[12];
  const float* b1e = (const float*)d_in[13];
  const float* W1  = (const float*)d_in[14];
  const float* b1  = (const float*)d_in[15];
  const float* W2  = (const float*)d_in[16];
  const float* b2  = (const float*)d_in[17];
  const float* g2h = (const float*)d_in[18];
  const float* b2h = (const float*)d_in[19];

  const size_t ND = (size_t)Nn * Dd;
  const size_t NH = (size_t)Nn * Hh;
  float* out_h3 = (float*)d_out;
  float* out_e2 = (float*)d_out + ND;

  float* ws    = (float*)d_ws;
  float* Qb    = ws;
  float* Kb    = Qb + ND;
  float* Vb    = Kb + ND;
  float* wV    = Vb + ND;
  float* zb    = wV + ND;
  float* h2raw = zb + NH;
  float* h2n   = h2raw + ND;
  float* X1    = h2n + ND;
  float* st    = X1 + 2 * ND;
  float* sum1 = st;        float* sq1 = st + 128;
  float* sumE = st + 256;  float* sqE = st + 384;
  float* sum2 = st + 512;  float* sq2 = st + 640;

  hipMemsetAsync(wV, 0, (ND + NH) * sizeof(float), stream);  // wV + z (contiguous)
  hipMemsetAsync(st, 0, 768 * sizeof(float), stream);        // all BN partial sums

  san_qkv<<<dim3(NBH, 3), 256, 0, stream>>>(h, Wq, Wk, Wv, Qb, Kb, Vb);
  san_edge<<<NBE, 256, 0, stream>>>(e, We, Qb, Kb, Vb, src, dst, out_e2, wV, zb,
                                    sumE, sqE);
  san_attnout<<<NBH, 256, 0, stream>>>(wV, zb, h, Wo, bo, h2raw, sum1, sq1);
  san_finalize<<<1, 128, 0, stream>>>(sum1, sq1, (float)Nn);
  san_finalize<<<1, 128, 0, stream>>>(sumE, sqE, (float)NEe);
  san_ffn1<<<NBH, 256, 0, stream>>>(h2raw, sum1, sq1, g1h, b1h, W1, b1, h2n, X1);
  san_ffn2<<<NBH, 256, 0, stream>>>(X1, W2, b2, h2n, out_h3, sum2, sq2);
  san_finalize<<<1, 128, 0, stream>>>(sum2, sq2, (float)Nn);
  san_bnapply<<<2048, 256, 0, stream>>>(out_h3, sum2, sq2, g2h, b2h, ND);
  san_bnapply<<<4096, 256, 0, stream>>>(out_e2, sumE, sqE, g1e, b1e, (size_t)NEe * Dd);
}